// FractalResoneticsSystem_26285199852121
// MI455X (gfx1250) — compile-verified
//
#include <hip/hip_runtime.h>
#include <hip/hip_bf16.h>

typedef __attribute__((ext_vector_type(2))) float v2f_;
typedef __attribute__((ext_vector_type(8))) float v8f_;

#define BATCH 65536
#define TPB 256
// LDS weight arena offsets (floats)
#define OFF_WL 12432
#define OFF_WI 12702
#define OFF_R  12711
#define SW_TOT 12797
// workspace layout (floats): [0..513] diff accumulators, [514..599] r0/r1/r2, [600..1367] Wf_eff 4x192
#define WS_R    514
#define WS_WF   600

__device__ __constant__ int c_nb[4]     = {2, 12, 72, 432};
__device__ __constant__ int c_w1[4]     = {0, 48, 336, 2064};
__device__ __constant__ int c_b1[4]     = {18, 156, 984, 5952};
__device__ __constant__ int c_w2[4]     = {24, 192, 1200, 7248};
__device__ __constant__ int c_b2[4]     = {42, 300, 1848, 11136};
__device__ __constant__ int c_prevn[4]  = {1, 2, 12, 72};
__device__ __constant__ int c_roff[4]   = {0, OFF_R + 0, OFF_R + 2, OFF_R + 14};
__device__ __constant__ int c_accb[4]   = {0, 1, 12, 83};

struct Params {
    const float* x; const float* Wl; const float* Wi;
    const float* W1[4]; const float* b1[4]; const float* W2[4]; const float* b2[4];
    float* ws; float* out;
};

__device__ inline void cp_lds(float* dst, const float* src, int n, int tid) {
    for (int i = tid; i < n; i += TPB) dst[i] = src[i];
}

// ---------------- prep: zero accumulators, row-sums of R, fold W_final ----------------
__global__ void prep_kernel(const float* __restrict__ R0, const float* __restrict__ R1,
                            const float* __restrict__ R2, const float* __restrict__ R3,
                            const float* __restrict__ Wf, const float* __restrict__ bf,
                            float* __restrict__ ws) {
    __shared__ float sr3[432];
    int tid = threadIdx.x;
    for (int i = tid; i < 514; i += TPB) ws[i] = 0.0f;   // must re-zero every launch
    if (tid < 2)        { float s = 0.f; for (int j = 0; j < 2;  ++j) s += R0[tid*2  + j];      ws[WS_R      + tid]      = s; }
    else if (tid < 14)  { int i = tid-2;  float s = 0.f; for (int j = 0; j < 12; ++j) s += R1[i*12 + j]; ws[WS_R + 2  + i] = s; }
    else if (tid < 86)  { int i = tid-14; float s = 0.f; for (int j = 0; j < 72; ++j) s += R2[i*72 + j]; ws[WS_R + 14 + i] = s; }
    for (int i = tid; i < 432; i += TPB) {
        float s = 0.f;
        for (int j = 0; j < 432; ++j) s += R3[i*432 + j];
        sr3[i] = s;
    }
    __syncthreads();
    // Wf_eff: rows g=0..2 = sum_i r3[i]*W_final[3i+g, :]; row 3 = b_final (bias via K=3 ones in A)
    for (int idx = tid; idx < 768; idx += TPB) {
        int g = idx / 192, k = idx % 192;
        float v = 0.f;
        if (k < 180) {
            if (g < 3) { float s = 0.f; for (int i = 0; i < 432; ++i) s += sr3[i] * Wf[(3*i + g)*180 + k]; v = s; }
            else v = bf[k];
        }
        ws[WS_WF + idx] = v;
    }
}

// ---------------- fused main kernel ----------------
__global__ void __launch_bounds__(TPB) fractal_main(Params p) {
    __shared__ float s_w[SW_TOT];
    __shared__ float s_diff[514];
    __shared__ float s_tile[TPB * 4];
    const int tid = threadIdx.x;

    for (int l = 0; l < 4; ++l) {
        int n = c_nb[l];
        cp_lds(s_w + c_w1[l], p.W1[l], n*9, tid);
        cp_lds(s_w + c_b1[l], p.b1[l], n*3, tid);
        cp_lds(s_w + c_w2[l], p.W2[l], n*9, tid);
        cp_lds(s_w + c_b2[l], p.b2[l], n*3, tid);
    }
    cp_lds(s_w + OFF_WL, p.Wl, 270, tid);
    cp_lds(s_w + OFF_WI, p.Wi, 9, tid);
    cp_lds(s_w + OFF_R,  p.ws + WS_R, 86, tid);
    for (int i = tid; i < 514; i += TPB) s_diff[i] = 0.0f;
    __syncthreads();

    // ---- seed: L = x@W_logic ; E = sin(2πL/3) ; I = E@W_int ; S = (L+E+I)/3
    const int b = blockIdx.x * TPB + tid;
    const float* xr = p.x + (size_t)b * 90;
    float L0 = 0.f, L1 = 0.f, L2 = 0.f;
    #pragma unroll 6
    for (int k = 0; k < 90; ++k) {
        float xv = xr[k];
        L0 = fmaf(xv, s_w[OFF_WL + k*3 + 0], L0);
        L1 = fmaf(xv, s_w[OFF_WL + k*3 + 1], L1);
        L2 = fmaf(xv, s_w[OFF_WL + k*3 + 2], L2);
    }
    const float w = 6.2831853071795864769f / 3.0f;
    float E0 = __sinf(w * L0), E1 = __sinf(w * L1), E2 = __sinf(w * L2);
    float S0 = (L0 + E0 + (E0*s_w[OFF_WI+0] + E1*s_w[OFF_WI+3] + E2*s_w[OFF_WI+6])) * (1.f/3.f);
    float S1 = (L1 + E1 + (E0*s_w[OFF_WI+1] + E1*s_w[OFF_WI+4] + E2*s_w[OFF_WI+7])) * (1.f/3.f);
    float S2 = (L2 + E2 + (E0*s_w[OFF_WI+2] + E1*s_w[OFF_WI+5] + E2*s_w[OFF_WI+8])) * (1.f/3.f);

    // ---- 4 fractal expander layers, state = 3 floats per sample
    for (int l = 0; l < 4; ++l) {
        const float* W1 = s_w + c_w1[l];
        const float* B1 = s_w + c_b1[l];
        const float* W2 = s_w + c_w2[l];
        const float* B2 = s_w + c_b2[l];
        const int n = c_nb[l], np = c_prevn[l], rb = c_roff[l], ab = c_accb[l];
        float T0 = 0.f, T1 = 0.f, T2 = 0.f;
        float p0 = 0.f, p1 = 0.f, p2 = 0.f;
        int ip = 0;
        for (int i = 0; i < n; ++i) {
            float c = (l == 0) ? 1.0f : s_w[rb + ip];
            if (++ip == np) ip = 0;
            float a0 = c*S0, a1 = c*S1, a2 = c*S2;
            const float* w1 = W1 + i*9; const float* w2 = W2 + i*9;
            float h0 = tanhf(fmaf(a0, w1[0], fmaf(a1, w1[3], fmaf(a2, w1[6], B1[i*3+0]))));
            float h1 = tanhf(fmaf(a0, w1[1], fmaf(a1, w1[4], fmaf(a2, w1[7], B1[i*3+1]))));
            float h2 = tanhf(fmaf(a0, w1[2], fmaf(a1, w1[5], fmaf(a2, w1[8], B1[i*3+2]))));
            float o0 = fmaf(h0, w2[0], fmaf(h1, w2[3], fmaf(h2, w2[6], B2[i*3+0])));
            float o1 = fmaf(h0, w2[1], fmaf(h1, w2[4], fmaf(h2, w2[7], B2[i*3+1])));
            float o2 = fmaf(h0, w2[2], fmaf(h1, w2[5], fmaf(h2, w2[8], B2[i*3+2])));
            T0 += o0; T1 += o1; T2 += o2;
            if (i > 0) {
                float ds = fabsf(o0-p0) + fabsf(o1-p1) + fabsf(o2-p2);
                #pragma unroll
                for (int m = 16; m >= 1; m >>= 1) ds += __shfl_xor(ds, m, 32);
                if ((tid & 31) == 0) atomicAdd(&s_diff[ab + i - 1], ds);
            }
            p0 = o0; p1 = o1; p2 = o2;
        }
        S0 = T0; S1 = T1; S2 = T2;
    }

    // ---- final: [B,4] @ [4,192] via V_WMMA_F32_16X16X4_F32 (K row 3 = ones -> adds b_final)
    s_tile[tid*4 + 0] = S0; s_tile[tid*4 + 1] = S1; s_tile[tid*4 + 2] = S2; s_tile[tid*4 + 3] = 1.0f;
    __syncthreads();

    const int wave = tid >> 5, lane = tid & 31;
    const int kp = lane >> 4, m = lane & 15;          // lane holds (K=2kp,2kp+1) of row M=m
    const float* wf = p.ws + WS_WF;
    v2f_ Bv[12];
    #pragma unroll
    for (int j = 0; j < 12; ++j) {                    // B: K x N = 4 x 16 per tile, mirrored layout
        Bv[j].x = wf[(kp ? 2 : 0)*192 + j*16 + m];
        Bv[j].y = wf[(kp ? 3 : 1)*192 + j*16 + m];
    }
    for (int tt = 0; tt < 2; ++tt) {
        int t = wave*2 + tt;                          // 16-sample tile within block
        int srow = (t*16 + m)*4;
        v2f_ A;
        A.x = s_tile[srow + (kp ? 2 : 0)];
        A.y = s_tile[srow + (kp ? 3 : 1)];
        #pragma unroll
        for (int j = 0; j < 12; ++j) {
            v8f_ Cz = {0.f,0.f,0.f,0.f,0.f,0.f,0.f,0.f};
            v8f_ D = __builtin_amdgcn_wmma_f32_16x16x4_f32(
                false, A, false, Bv[j], (short)0, Cz, false, false);
            int ncol = j*16 + m;
            if (ncol < 180) {
                size_t base = (size_t)blockIdx.x * TPB + t*16 + (kp ? 8 : 0);
                #pragma unroll
                for (int v = 0; v < 8; ++v)           // D: vgpr v -> row M=v (+8 for lanes 16..31)
                    p.out[(base + v)*180 + ncol] = D[v];
            }
        }
    }

    // ---- flush per-block diff partial sums
    for (int i = tid; i < 514; i += TPB) atomicAdd(&p.ws[i], s_diff[i]);
}

// ---------------- stability finalize ----------------
__global__ void stab_kernel(const float* __restrict__ ws, float* __restrict__ out) {
    int idx = blockIdx.x * blockDim.x + threadIdx.x;
    if (idx >= 518) return;
    int l, local;
    if (idx < 2)       { l = 0; local = idx; }
    else if (idx < 14) { l = 1; local = idx - 2; }
    else if (idx < 86) { l = 2; local = idx - 14; }
    else               { l = 3; local = idx - 86; }
    const int accb[4] = {0, 1, 12, 83};
    float v;
    if (local == 0) v = 1.0f;
    else {
        float d = ws[accb[l] + local - 1] * (1.0f / (65536.0f * 3.0f));
        v = fmaxf(0.0f, 1.0f - d);
    }
    out[(size_t)BATCH * 180 + idx] = v;
}

extern "C" void kernel_launch(void* const* d_in, const int* in_sizes, int n_in,
                              void* d_out, int out_size, void* d_ws, size_t ws_size,
                              hipStream_t stream) {
    (void)in_sizes; (void)n_in; (void)out_size; (void)ws_size;
    float* ws = (float*)d_ws;
    Params p;
    p.x  = (const float*)d_in[0];
    p.Wl = (const float*)d_in[1];
    p.Wi = (const float*)d_in[2];
    for (int l = 0; l < 4; ++l) {
        p.W1[l] = (const float*)d_in[3 + l*5 + 0];
        p.b1[l] = (const float*)d_in[3 + l*5 + 1];
        p.W2[l] = (const float*)d_in[3 + l*5 + 2];
        p.b2[l] = (const float*)d_in[3 + l*5 + 3];
    }
    const float* R0 = (const float*)d_in[7];
    const float* R1 = (const float*)d_in[12];
    const float* R2 = (const float*)d_in[17];
    const float* R3 = (const float*)d_in[22];
    const float* Wf = (const float*)d_in[23];
    const float* bf = (const float*)d_in[24];
    p.ws = ws;
    p.out = (float*)d_out;

    prep_kernel<<<1, TPB, 0, stream>>>(R0, R1, R2, R3, Wf, bf, ws);
    fractal_main<<<BATCH / TPB, TPB, 0, stream>>>(p);
    stab_kernel<<<(518 + TPB - 1) / TPB, TPB, 0, stream>>>(ws, (float*)d_out);
}